// SASA_Layer_8881992368840
// MI455X (gfx1250) — compile-verified
//
#include <hip/hip_runtime.h>
#include <hip/hip_bf16.h>

// SASA layer: B=4, C=128, H=W=64, K=7, NH=8, DH=16, PAD=3
// Stage 1 (WMMA): q = wq*x+bq on 64x64; k,v = w*pad(x)+b on the padded 70x70
//   grid (border neighbors correctly carry the bias, matching the reference's
//   pad->conv->unfold order).  fp32 GEMM, 128-deep K -> V_WMMA_F32_16X16X4_F32.
//   Block = one 16-pixel N-tile staged in LDS, 8 waves = 8 M-tiles.
// Stage 2 (VALU): per-pixel 7x7 attention (logits, softmax, weighted V).

typedef __attribute__((ext_vector_type(2))) float v2f;
typedef __attribute__((ext_vector_type(8))) float v8f;

#define BB  4
#define CC  128
#define HH  64
#define WW  64
#define HP  70
#define WP  70
#define NPIX   (HH * WW)       // 4096
#define NPAD   (HP * WP)       // 4900
#define NHEAD  8
#define DH     16
#define KS     7
#define LDS_STRIDE 17          // 16 + 1 pad: avoids 2-way bank conflict

// ---------------------------------------------------------------------------
// 1x1 conv as GEMM:  Out[o, p] = sum_c Wm[o,c] * X[c,p] + bias[o]
// One block = one (batch, 16-pixel tile); wave w = out-channels [16w, 16w+16).
// X tile (128 x 16 fp32) cooperatively loaded to LDS once, reused by 8 waves.
// padded==1: pixels index the 70x70 padded grid; x loads are clamped+masked
// (unconditional vmem + v_cndmask, no exec-mask predication).
// ---------------------------------------------------------------------------
__global__ void __launch_bounds__(256)
conv1x1_wmma(const float* __restrict__ X,    // [B,128,64,64]
             const float* __restrict__ Wm,   // [128,128] row-major (o,c)
             const float* __restrict__ bias, // [128]
             float* __restrict__ Out,        // [B,128,npix]
             int npix, int padded)
{
    __shared__ float xs[CC * LDS_STRIDE];    // 8.5 KB

    const int tid  = threadIdx.x;
    const int lane = tid & 31;
    const int wave = tid >> 5;               // M tile 0..7
    const int tilesN = (npix + 15) >> 4;

    const int b  = blockIdx.x / tilesN;
    const int nt = blockIdx.x % tilesN;
    const int n0 = nt * 16;

    // ---- cooperative X-tile fill: thread -> pixel p = tid%16, k = (tid/16)*8 ..+7
    {
        const int p  = tid & 15;
        const int kb = (tid >> 4) * 8;       // 16 groups x 8 k's = 128
        int pix = n0 + p;
        if (pix >= npix) pix = npix - 1;     // clamp (stores masked later)
        float mask = 1.0f;
        int src;
        if (padded) {
            const int i  = pix / WP, j = pix % WP;
            int xi = i - 3, xj = j - 3;
            const bool inimg = (xi >= 0) & (xi < HH) & (xj >= 0) & (xj < WW);
            mask = inimg ? 1.0f : 0.0f;
            xi = min(max(xi, 0), HH - 1);    // always-legal address
            xj = min(max(xj, 0), WW - 1);
            src = xi * WW + xj;
        } else {
            src = pix;
        }
        const float* Xp = X + (size_t)b * CC * NPIX + src;
        #pragma unroll
        for (int kk = 0; kk < 8; ++kk)
            xs[(kb + kk) * LDS_STRIDE + p] = mask * Xp[(size_t)(kb + kk) * NPIX];
    }
    __syncthreads();

    const int half = lane >> 4;              // 0|1 (K pair select)
    const int l16  = lane & 15;
    const int m0   = wave * 16;

    const float* wrow = Wm + (size_t)(m0 + l16) * CC;

    v8f acc = {};
    #pragma unroll
    for (int k0 = 0; k0 < CC; k0 += 4) {
        const int ka = k0 + 2 * half;
        // A: 16x4 f32 -> lane holds Wm[m0+l16, ka..ka+1] (one b64 load)
        v2f a;
        a.x = wrow[ka];
        a.y = wrow[ka + 1];
        // B: 4x16 f32 -> lane holds X[ka..ka+1, n0+l16] from LDS
        v2f bv;
        bv.x = xs[ka * LDS_STRIDE + l16];
        bv.y = xs[(ka + 1) * LDS_STRIDE + l16];
        acc = __builtin_amdgcn_wmma_f32_16x16x4_f32(
                  false, a, false, bv, (short)0, acc, false, false);
    }

    // C/D layout: lane L, vreg r -> M = r + 8*(L/16), N = L%16
    float* Ob = Out + (size_t)b * CC * npix;
    const int outpix = n0 + l16;
    if (outpix < npix) {
        #pragma unroll
        for (int r = 0; r < 8; ++r) {
            const int o = m0 + r + 8 * half;
            Ob[(size_t)o * npix + outpix] = acc[r] + bias[o];
        }
    }
}

// ---------------------------------------------------------------------------
// Per-pixel 7x7 local attention.  One thread = one (b, head, h, w).
// Block = 64 threads spanning a W row -> k/v neighborhood rows are shared.
// ---------------------------------------------------------------------------
__global__ void __launch_bounds__(64)
sasa_attn(const float* __restrict__ Q,     // [B,128,64,64]
          const float* __restrict__ Kp,    // [B,128,70,70]
          const float* __restrict__ Vp,    // [B,128,70,70]
          const float* __restrict__ pos_h, // [8,7]
          const float* __restrict__ pos_w, // [8,7]
          float* __restrict__ Out)         // [B,128,64,64]
{
    const int w   = threadIdx.x;           // 0..63
    const int blk = blockIdx.x;            // b*512 + n*64 + h
    const int h = blk & 63;
    const int n = (blk >> 6) & 7;
    const int b = blk >> 9;

    const size_t qbase = (((size_t)b * CC + n * DH) * HH + h) * WW + w;

    float q[DH];
    #pragma unroll
    for (int d = 0; d < DH; ++d) q[d] = Q[qbase + (size_t)d * NPIX];

    float ph[KS], pw[KS];
    #pragma unroll
    for (int t = 0; t < KS; ++t) {
        ph[t] = pos_h[n * KS + t];
        pw[t] = pos_w[n * KS + t];
    }

    const size_t kbase = ((size_t)b * CC + n * DH) * NPAD;

    float logits[KS * KS];
    float m = -3.4e38f;
    #pragma unroll
    for (int kh = 0; kh < KS; ++kh) {
        #pragma unroll
        for (int kw = 0; kw < KS; ++kw) {
            const size_t off = kbase + (size_t)(h + kh) * WP + (w + kw);
            float s = 0.0f;
            #pragma unroll
            for (int d = 0; d < DH; ++d)
                s += q[d] * Kp[off + (size_t)d * NPAD];
            s += ph[kh] + pw[kw];
            logits[kh * KS + kw] = s;
            m = fmaxf(m, s);
        }
    }

    float sum = 0.0f;
    #pragma unroll
    for (int t = 0; t < KS * KS; ++t) {
        logits[t] = __expf(logits[t] - m);
        sum += logits[t];
    }
    const float inv = 1.0f / sum;

    float acc[DH];
    #pragma unroll
    for (int d = 0; d < DH; ++d) acc[d] = 0.0f;

    #pragma unroll
    for (int kh = 0; kh < KS; ++kh) {
        #pragma unroll
        for (int kw = 0; kw < KS; ++kw) {
            const float wgt = logits[kh * KS + kw];
            const size_t off = kbase + (size_t)(h + kh) * WP + (w + kw);
            #pragma unroll
            for (int d = 0; d < DH; ++d)
                acc[d] += wgt * Vp[off + (size_t)d * NPAD];
        }
    }

    #pragma unroll
    for (int d = 0; d < DH; ++d)
        Out[qbase + (size_t)d * NPIX] = acc[d] * inv;
}

extern "C" void kernel_launch(void* const* d_in, const int* in_sizes, int n_in,
                              void* d_out, int out_size, void* d_ws, size_t ws_size,
                              hipStream_t stream) {
    const float* x     = (const float*)d_in[0];
    const float* wq    = (const float*)d_in[1];
    const float* bq    = (const float*)d_in[2];
    const float* wk    = (const float*)d_in[3];
    const float* bk    = (const float*)d_in[4];
    const float* wv    = (const float*)d_in[5];
    const float* bv    = (const float*)d_in[6];
    const float* pos_h = (const float*)d_in[7];
    const float* pos_w = (const float*)d_in[8];
    float* out = (float*)d_out;

    float* qbuf = (float*)d_ws;                                   // 4*128*4096
    float* kbuf = qbuf + (size_t)BB * CC * NPIX;                  // 4*128*4900
    float* vbuf = kbuf + (size_t)BB * CC * NPAD;

    // Q: one block per (batch, 16-pixel tile): 4 * 256 = 1024 blocks
    conv1x1_wmma<<<1024, 256, 0, stream>>>(x, wq, bq, qbuf, NPIX, 0);
    // K,V on padded grid: 4 * ceil(4900/16)=307 -> 1228 blocks each
    conv1x1_wmma<<<1228, 256, 0, stream>>>(x, wk, bk, kbuf, NPAD, 1);
    conv1x1_wmma<<<1228, 256, 0, stream>>>(x, wv, bv, vbuf, NPAD, 1);
    // Attention: B*NH*H = 2048 blocks of 64 threads (one W row each)
    sasa_attn<<<2048, 64, 0, stream>>>(qbuf, kbuf, vbuf, pos_h, pos_w, out);
}